// Flinear_65455301591497
// MI455X (gfx1250) — compile-verified
//
#include <hip/hip_runtime.h>

typedef __attribute__((ext_vector_type(2))) float v2f;
typedef __attribute__((ext_vector_type(4))) float v4f;
typedef __attribute__((ext_vector_type(8))) float v8f;

#define NROWS   262144
#define IN_DIM  128
#define OUT_DIM 256
#define BASIS   7

// One wave (32 lanes) computes a 16-row tile:
//   D = A x B, A = W_s^T padded to 16x4 (M=basis, K), B = x-tile^T 4x16 (K, N=row)
//   -> D[m,n] = h[row n, basis m]; lanes 0..15 hold the full h-vector of one row.
__global__ __launch_bounds__(256) void fourier_feature_kernel(
    const float* __restrict__ x,   // [NROWS, IN_DIM]
    const float* __restrict__ Ws,  // [IN_DIM, BASIS]
    const float* __restrict__ bs,  // [BASIS]
    const float* __restrict__ Wm,  // [2*BASIS, 1]
    const float* __restrict__ bm,  // [1]
    float* __restrict__ out)       // [NROWS, OUT_DIM]
{
    // W_s transposed + zero-padded to 16 basis rows: Wt[m][k] = W_s[k][m]
    __shared__ float Wt[16][IN_DIM];   // 8 KB

    const int tid = threadIdx.x;
    #pragma unroll
    for (int i = tid; i < 16 * IN_DIM; i += 256) {
        const int m = i / IN_DIM;
        const int k = i % IN_DIM;
        Wt[m][k] = (m < BASIS) ? Ws[k * BASIS + m] : 0.0f;
    }
    __syncthreads();

    const int wave = tid >> 5;          // wave32
    const int lane = tid & 31;
    const int row_base = blockIdx.x * 128 + wave * 16;

    const int nIdx  = lane & 15;        // row within tile (B: N=lane%16) / basis (A: M=lane%16)
    const int kHalf = (lane >> 4) * 2;  // K sub-offset: lanes 16..31 handle K+2,K+3

    const float* __restrict__ xrow = x + (size_t)(row_base + nIdx) * IN_DIM + kHalf;
    const float* __restrict__ wrow = &Wt[nIdx][kHalf];

    v8f c = {};
    #pragma unroll
    for (int kb = 0; kb < IN_DIM; kb += 4) {
        v2f a = *(const v2f*)(wrow + kb);   // A fragment: W_s^T[m, kb+kHalf .. +1] (ds_load_b64)
        v2f b = *(const v2f*)(xrow + kb);   // B fragment: x[row, kb+kHalf .. +1] (global_load_b64)
        // 8 args: (neg_a, A, neg_b, B, c_mod, C, reuse_a, reuse_b)
        c = __builtin_amdgcn_wmma_f32_16x16x4_f32(
                false, a, false, b, (short)0, c, false, false);
    }

    // Epilogue: lanes 0..15 hold h[row = lane][basis m] in c[m], m = 0..6 (rest is padding).
    float y = bm[0];
    #pragma unroll
    for (int m = 0; m < BASIS; ++m) {
        const float h = c[m] + bs[m];                       // bs/Wm uniform -> scalar loads
        y += __sinf(h) * Wm[m] + __cosf(h) * Wm[BASIS + m];
    }

    // Broadcast y per row and tile it across 256 output columns (coalesced b128 stores).
    float* __restrict__ orow = out + (size_t)row_base * OUT_DIM;
    #pragma unroll
    for (int r = 0; r < 16; ++r) {
        const float yv = __shfl(y, r, 32);
        const v4f v = { yv, yv, yv, yv };
        *(v4f*)(orow + (size_t)r * OUT_DIM + lane * 4)       = v;
        *(v4f*)(orow + (size_t)r * OUT_DIM + 128 + lane * 4) = v;
    }
}

extern "C" void kernel_launch(void* const* d_in, const int* in_sizes, int n_in,
                              void* d_out, int out_size, void* d_ws, size_t ws_size,
                              hipStream_t stream) {
    (void)in_sizes; (void)n_in; (void)out_size; (void)d_ws; (void)ws_size;
    const float* x  = (const float*)d_in[0];
    const float* Ws = (const float*)d_in[1];
    const float* bs = (const float*)d_in[2];
    const float* Wm = (const float*)d_in[3];
    const float* bm = (const float*)d_in[4];
    float* out = (float*)d_out;

    dim3 grid(NROWS / 128);   // 2048 blocks, 8 waves x 16 rows each
    dim3 block(256);
    fourier_feature_kernel<<<grid, block, 0, stream>>>(x, Ws, bs, Wm, bm, out);
}